// InteractionNetwork_13082470383740
// MI455X (gfx1250) — compile-verified
//
#include <hip/hip_runtime.h>
#include <math.h>

// Problem constants (from reference)
#define B_      4
#define N_      2048     // 2^11
#define E_      8192     // 2^13
#define REL_HID 150
#define OBJ_HID 100
#define KPAD    152      // 150 rounded up to multiple of 4
#define NPAD    160      // padded output width (10 tiles of 16)
#define HPAD    160      // LDS row pitch for h1

typedef __attribute__((ext_vector_type(2))) float v2f;
typedef __attribute__((ext_vector_type(4))) float v4f;
typedef __attribute__((ext_vector_type(8))) float v8f;

// ---------------------------------------------------------------------------
// Phase 1: recover gather indices from the dense one-hot incidence matrices.
// onehot is [B, N, E]; column e has exactly one 1 at row n.  One coalesced
// non-temporal streaming pass (this is the HBM-bound part: 2 x 256 MB,
// ~22 us at 23.3 TB/s -- the floor for this whole problem).
// ---------------------------------------------------------------------------
__global__ void extract_idx_kernel(const float* __restrict__ onehot,
                                   int* __restrict__ idx) {
  const long long total4 = (long long)B_ * N_ * E_ / 4;
  const long long stride = (long long)gridDim.x * blockDim.x;
  for (long long i4 = blockIdx.x * (long long)blockDim.x + threadIdx.x;
       i4 < total4; i4 += stride) {
    const v4f* p = (const v4f*)onehot + i4;
    v4f v = __builtin_nontemporal_load(p);
#pragma unroll
    for (int c = 0; c < 4; ++c) {
      if (v[c] > 0.5f) {
        long long i = i4 * 4 + c;           // i = ((b*N)+n)*E + e
        int e = (int)(i & (E_ - 1));
        long long bn = i >> 13;             // / E_
        int n = (int)(bn & (N_ - 1));
        int b = (int)(bn >> 11);            // / N_
        idx[b * E_ + e] = n;
      }
    }
  }
}

__global__ void zero_kernel(float* __restrict__ p, int n) {
  int i = blockIdx.x * blockDim.x + threadIdx.x;
  if (i < n) p[i] = 0.f;
}

// ---------------------------------------------------------------------------
// Pre-pad relational weights so the WMMA inner loop is branchless:
//   W2p: [KPAD/2][NPAD][2]  pairs {W2[2kp][n], W2[2kp+1][n]}, zero padded
//   b2p: [NPAD]  zero padded bias
//   W3p: [NPAD]  zero padded layer-3 weights
// ---------------------------------------------------------------------------
__global__ void pad_weights_kernel(const float* __restrict__ W2,
                                   const float* __restrict__ b2,
                                   const float* __restrict__ W3,
                                   float* __restrict__ W2p,
                                   float* __restrict__ b2p,
                                   float* __restrict__ W3p) {
  const int t = blockIdx.x * blockDim.x + threadIdx.x;
  const int npairs = (KPAD / 2) * NPAD;          // 76 * 160
  if (t < npairs) {
    const int kp = t / NPAD;
    const int n  = t - kp * NPAD;
    const int k0 = 2 * kp, k1 = 2 * kp + 1;
    W2p[t * 2 + 0] = (k0 < REL_HID && n < REL_HID) ? W2[k0 * REL_HID + n] : 0.f;
    W2p[t * 2 + 1] = (k1 < REL_HID && n < REL_HID) ? W2[k1 * REL_HID + n] : 0.f;
  }
  if (t < NPAD) {
    b2p[t] = (t < REL_HID) ? b2[t] : 0.f;
    W3p[t] = (t < REL_HID) ? W3[t] : 0.f;
  }
}

// ---------------------------------------------------------------------------
// Relational MLP over edges: x[7] -> relu(150) -> relu(150) -> sigmoid(1).
// One wave32 per tile of 16 edges.  Layer 2 uses V_WMMA_F32_16X16X4_F32:
// A (16x4 f32): lanes 0-15 hold M=lane, K={k0,k0+1}; lanes 16-31 K={k0+2,k0+3}
// B (4x16 f32): lane holds col n=lane&15; low lanes rows {k0,k0+1}, high {k0+2,k0+3}
// C/D (16x16 f32): c[r] = row r (lanes 0-15) / row r+8 (lanes 16-31), col = lane&15
// Layer 3 (150->1) + sigmoid fused via 16-lane xor-shuffle reduction.
// FINAL=0: scatter-add eff into eff_rcv[b*N + rcv] (atomic f32).
// FINAL=1: write edge weights to out[b*E + e].
// ---------------------------------------------------------------------------
template <int FINAL>
__global__ __launch_bounds__(128)
void relational_kernel(const float* __restrict__ obj,      // [B,N,3]
                       const int* __restrict__ snd_idx,    // [B*E]
                       const int* __restrict__ rcv_idx,    // [B*E]
                       const float* __restrict__ rel,      // [B,1,E]
                       const float* __restrict__ W1, const float* __restrict__ b1,
                       const float* __restrict__ W2p,      // padded pairs (see above)
                       const float* __restrict__ b2p,
                       const float* __restrict__ W3p,
                       const float* __restrict__ b3,
                       float* __restrict__ dst) {
  __shared__ float h1[4][16][HPAD];   // layer-1 activations, zero-padded K
  __shared__ float xin[4][16][8];     // [snd(3), rcv(3), rel(1), 0]

  const int lane  = threadIdx.x & 31;
  const int w     = threadIdx.x >> 5;
  const int tile  = blockIdx.x * 4 + w;
  const int g0    = tile * 16;        // first edge of tile
  const int b     = g0 >> 13;         // / E_ (tiles never straddle batches)

  // ---- gather inputs for 16 edges ----
  if (lane < 16) {
    const int g = g0 + lane;
    const int e = g & (E_ - 1);
    const int sn = snd_idx[g];
    const int rn = rcv_idx[g];
    const float* po = obj + (long long)b * N_ * 3;
    xin[w][lane][0] = po[sn * 3 + 0];
    xin[w][lane][1] = po[sn * 3 + 1];
    xin[w][lane][2] = po[sn * 3 + 2];
    xin[w][lane][3] = po[rn * 3 + 0];
    xin[w][lane][4] = po[rn * 3 + 1];
    xin[w][lane][5] = po[rn * 3 + 2];
    xin[w][lane][6] = rel[(long long)b * E_ + e];
    xin[w][lane][7] = 0.f;
  }
  __syncthreads();

  // ---- layer 1: x[16x7] @ W1[7x150] + b1, relu; K-pad to HPAD with zeros ----
  for (int j = lane; j < HPAD; j += 32) {
    float wc[7];
    float bias = 0.f;
    const bool valid = (j < REL_HID);
    if (valid) {
#pragma unroll
      for (int i = 0; i < 7; ++i) wc[i] = W1[i * REL_HID + j];
      bias = b1[j];
    } else {
#pragma unroll
      for (int i = 0; i < 7; ++i) wc[i] = 0.f;
    }
    for (int m = 0; m < 16; ++m) {
      float acc = bias;
#pragma unroll
      for (int i = 0; i < 7; ++i) acc += xin[w][m][i] * wc[i];
      h1[w][m][j] = valid ? fmaxf(acc, 0.f) : 0.f;
    }
  }
  __syncthreads();

  // ---- layer 2 via WMMA f32 16x16x4, fused layer-3 epilogue (branchless) ----
  const int laneN = lane & 15;
  const int hiK   = (lane >> 4) * 2;       // 0 for low half, 2 for high half
  const v2f* W2pv = (const v2f*)W2p;       // [KPAD/2][NPAD] of K-pairs
  const float* h1row = &h1[w][laneN][0];   // 8-byte aligned (HPAD even)

  float eacc[8];
#pragma unroll
  for (int r = 0; r < 8; ++r) eacc[r] = 0.f;

  for (int nt = 0; nt < 10; ++nt) {        // 150 outputs -> 10 tiles of 16
    const int n = nt * 16 + laneN;
    v8f c = {};
#pragma unroll 2
    for (int kc = 0; kc < KPAD / 4; ++kc) {            // 38 K-steps of 4
      const v2f a  = *(const v2f*)(h1row + kc * 4 + hiK);   // ds_load_b64
      const v2f bf = W2pv[(kc * 2 + (hiK >> 1)) * NPAD + n]; // global_load_b64
      c = __builtin_amdgcn_wmma_f32_16x16x4_f32(
          /*neg_a=*/false, a, /*neg_b=*/false, bf,
          /*c_mod=*/(short)0, c, /*reuse_a=*/false, /*reuse_b=*/false);
    }
    const float bb = b2p[n];
    const float w3 = W3p[n];
#pragma unroll
    for (int r = 0; r < 8; ++r) {
      float h = fmaxf(c[r] + bb, 0.f);     // relu(h2)
      eacc[r] += h * w3;                   // layer-3 partial dot
    }
  }
  // reduce over the 16 lanes holding different hidden-unit columns
#pragma unroll
  for (int off = 1; off < 16; off <<= 1) {
#pragma unroll
    for (int r = 0; r < 8; ++r) eacc[r] += __shfl_xor(eacc[r], off, 32);
  }
  if (laneN == 0) {
    const int rowBase = (lane >> 4) * 8;  // lane 0 -> edges 0-7, lane 16 -> 8-15
#pragma unroll
    for (int r = 0; r < 8; ++r) {
      const int g = g0 + rowBase + r;
      const float z = eacc[r] + b3[0];
      const float eff = 1.f / (1.f + __expf(-z));
      if (FINAL) {
        dst[g] = eff;                                         // edge weights
      } else {
        atomicAdd(&dst[(long long)b * N_ + rcv_idx[g]], eff); // scatter-add
      }
    }
  }
}

// ---------------------------------------------------------------------------
// Object MLP: [obj(3), eff_rcv(1)] -> relu(100) -> 3.  One thread per node.
// ---------------------------------------------------------------------------
__global__ void object_kernel(const float* __restrict__ obj,
                              const float* __restrict__ effr,
                              const float* __restrict__ W1, const float* __restrict__ b1,
                              const float* __restrict__ W2, const float* __restrict__ b2,
                              float* __restrict__ out) {
  const int i = blockIdx.x * blockDim.x + threadIdx.x;
  if (i >= B_ * N_) return;
  const float a0 = obj[i * 3 + 0], a1 = obj[i * 3 + 1], a2 = obj[i * 3 + 2];
  const float a3 = effr[i];
  float o0 = b2[0], o1 = b2[1], o2 = b2[2];
  for (int j = 0; j < OBJ_HID; ++j) {
    float h = b1[j] + a0 * W1[0 * OBJ_HID + j] + a1 * W1[1 * OBJ_HID + j] +
              a2 * W1[2 * OBJ_HID + j] + a3 * W1[3 * OBJ_HID + j];
    h = fmaxf(h, 0.f);
    o0 += h * W2[j * 3 + 0];
    o1 += h * W2[j * 3 + 1];
    o2 += h * W2[j * 3 + 2];
  }
  out[i * 3 + 0] = o0;
  out[i * 3 + 1] = o1;
  out[i * 3 + 2] = o2;
}

// ---------------------------------------------------------------------------
extern "C" void kernel_launch(void* const* d_in, const int* in_sizes, int n_in,
                              void* d_out, int out_size, void* d_ws, size_t ws_size,
                              hipStream_t stream) {
  const float* objects  = (const float*)d_in[0];
  const float* snd_rel  = (const float*)d_in[1];
  const float* rcv_rel  = (const float*)d_in[2];
  const float* rel_info = (const float*)d_in[3];
  const float* rW1 = (const float*)d_in[4];
  const float* rb1 = (const float*)d_in[5];
  const float* rW2 = (const float*)d_in[6];
  const float* rb2 = (const float*)d_in[7];
  const float* rW3 = (const float*)d_in[8];
  const float* rb3 = (const float*)d_in[9];
  const float* oW1 = (const float*)d_in[10];
  const float* ob1 = (const float*)d_in[11];
  const float* oW2 = (const float*)d_in[12];
  const float* ob2 = (const float*)d_in[13];
  float* out = (float*)d_out;

  // workspace layout
  int*   snd_idx = (int*)d_ws;
  int*   rcv_idx = snd_idx + B_ * E_;
  float* effr    = (float*)(rcv_idx + B_ * E_);       // [B*N]
  float* obj2    = effr + B_ * N_;                    // [B*N*3]
  float* W2p     = obj2 + B_ * N_ * 3;                // [KPAD/2 * NPAD * 2]
  float* b2p     = W2p + (KPAD / 2) * NPAD * 2;       // [NPAD]
  float* W3p     = b2p + NPAD;                        // [NPAD]

  // 1) recover gather indices (HBM-bound streaming pass, 512 MB total)
  extract_idx_kernel<<<8192, 256, 0, stream>>>(snd_rel, snd_idx);
  extract_idx_kernel<<<8192, 256, 0, stream>>>(rcv_rel, rcv_idx);

  // 2) zero scatter target + pad weights (tiny, overlaps with nothing big)
  const int bn = B_ * N_;
  zero_kernel<<<(bn + 255) / 256, 256, 0, stream>>>(effr, bn);
  const int npairs = (KPAD / 2) * NPAD;
  pad_weights_kernel<<<(npairs + 255) / 256, 256, 0, stream>>>(
      rW2, rb2, rW3, W2p, b2p, W3p);

  // 3) recurrence pass: relational MLP -> scatter-add effects
  const int tiles = B_ * E_ / 16;   // 2048 tiles, 4 waves/block
  relational_kernel<0><<<tiles / 4, 128, 0, stream>>>(
      objects, snd_idx, rcv_idx, rel_info, rW1, rb1, W2p, b2p, W3p, rb3, effr);

  // 4) object MLP -> updated objects
  object_kernel<<<(bn + 255) / 256, 256, 0, stream>>>(
      objects, effr, oW1, ob1, oW2, ob2, obj2);

  // 5) final relational pass -> edge weights (sigmoid) into d_out
  relational_kernel<1><<<tiles / 4, 128, 0, stream>>>(
      obj2, snd_idx, rcv_idx, rel_info, rW1, rb1, W2p, b2p, W3p, rb3, out);
}